// PSPGnnDGL_2602750181865
// MI455X (gfx1250) — compile-verified
//
#include <hip/hip_runtime.h>
#include <hip/hip_bf16.h>
#include <math.h>

// ---------------------------------------------------------------------------
// PSP GNN (EGAT, 3 layers) forward for MI455X / gfx1250 (wave32, WMMA).
// All GEMMs run through one fp32 WMMA kernel (V_WMMA_F32_16X16X4_F32).
// ---------------------------------------------------------------------------

#define HIDN 64
#define NHEADS 2
#define HOUT 128          // NHEADS*HIDN
#define INDIM 10
#define FDIM 266          // INDIM + 4*HIDN
#define NPG 6250
#define BGRAPH 8
#define NLAYERS 3
#define NEGSLOPE 0.2f

typedef float v2f __attribute__((ext_vector_type(2)));
typedef float v8f __attribute__((ext_vector_type(8)));

__device__ __forceinline__ unsigned psp_enc(float f) {
    unsigned b = __float_as_uint(f);
    return (b & 0x80000000u) ? ~b : (b | 0x80000000u);
}
__device__ __forceinline__ float psp_dec(unsigned u) {
    unsigned b = (u & 0x80000000u) ? (u & 0x7FFFFFFFu) : ~u;
    return __uint_as_float(b);
}

// ---------------------------------------------------------------------------
// Generic fp32 GEMM: C[M x Nc] = A[M x K] * B[K x Nc]  (+bias)(+relu)(+resid)
// One wave per 16x16 output tile; K consumed 4 at a time via WMMA f32 16x16x4.
// A 16x4 layout: lanes 0-15 hold (K=k,k+1) of row M=lane; lanes 16-31 (K=k+2,k+3).
// B 4x16 layout: VGPR0 lanes0-15=K row k, lanes16-31=row k+2; VGPR1 = k+1 / k+3.
// C/D: VGPR r -> M = r (lanes 0-15) / r+8 (lanes 16-31), N = lane&15.
// ---------------------------------------------------------------------------
__global__ void psp_gemm_wmma(const float* __restrict__ A, int lda,
                              const float* __restrict__ Bm, int ldb,
                              float* __restrict__ C, int ldc,
                              int M, int K, int Nc,
                              const float* __restrict__ bias, int relu,
                              const float* __restrict__ resid, int rld) {
    const int wave = blockIdx.x * (blockDim.x >> 5) + (threadIdx.x >> 5);
    const int ntn = Nc >> 4;
    const int ntm = (M + 15) >> 4;
    if (wave >= ntm * ntn) return;                // wave-uniform: EXEC stays full
    const int m0 = (wave / ntn) << 4;
    const int n0 = (wave % ntn) << 4;
    const int lane = threadIdx.x & 31;
    const int half = lane >> 4;
    const int l16 = lane & 15;
    int arow = m0 + l16; arow = arow < M ? arow : (M - 1);
    const int bcol = n0 + l16;

    v8f acc = {};
    const int K4 = K & ~3;
    for (int k = 0; k < K4; k += 4) {
        v2f a, b;
        const float* ap = A + (size_t)arow * lda + k + (half << 1);
        a.x = ap[0];
        a.y = ap[1];
        const float* bp = Bm + (size_t)(k + (half << 1)) * ldb + bcol;
        b.x = bp[0];
        b.y = bp[ldb];
        acc = __builtin_amdgcn_wmma_f32_16x16x4_f32(false, a, false, b,
                                                    (short)0, acc, false, false);
    }
    for (int k = K4; k < K; ++k) {               // K tail (feat encoder, K=10)
        float bv = Bm[(size_t)k * ldb + bcol];
#pragma unroll
        for (int r = 0; r < 8; ++r) {
            int rr = m0 + r + (half << 3); rr = rr < M ? rr : (M - 1);
            acc[r] += A[(size_t)rr * lda + k] * bv;
        }
    }
#pragma unroll
    for (int r = 0; r < 8; ++r) {
        int row = m0 + r + (half << 3);
        if (row < M) {
            float v = acc[r];
            if (bias)  v += bias[bcol];
            if (relu)  v = v > 0.f ? v : 0.f;
            if (resid) v += resid[(size_t)row * rld + bcol];
            C[(size_t)row * ldc + bcol] = v;
        }
    }
}

// features[:, 0:10] = feat
__global__ void psp_copy_feat(const float* __restrict__ feat,
                              float* __restrict__ features, int N) {
    int total = N * INDIM;
    for (int i = blockIdx.x * blockDim.x + threadIdx.x; i < total;
         i += gridDim.x * blockDim.x) {
        int n = i / INDIM, c = i - n * INDIM;
        features[(size_t)n * FDIM + c] = feat[i];
    }
}

// emb[e,k] = etype_emb[etype[e],k] + rid_emb[rid[e],k] + att_rc@rc_W + rc_b
//            + att_rp@rp_W + rp_b
__global__ void psp_edge_emb(const float* __restrict__ att_rc,
                             const float* __restrict__ att_rp,
                             const float* __restrict__ etype_emb,
                             const float* __restrict__ rid_emb,
                             const float* __restrict__ rc_W, const float* __restrict__ rc_b,
                             const float* __restrict__ rp_W, const float* __restrict__ rp_b,
                             const int* __restrict__ etype, const int* __restrict__ rid,
                             float* __restrict__ emb, int E) {
    int total = E * HIDN;
    for (int i = blockIdx.x * blockDim.x + threadIdx.x; i < total;
         i += gridDim.x * blockDim.x) {
        int e = i >> 6, k = i & 63;
        float v = etype_emb[etype[e] * HIDN + k] + rid_emb[rid[e] * HIDN + k]
                + att_rc[e * 2 + 0] * rc_W[k] + att_rc[e * 2 + 1] * rc_W[HIDN + k] + rc_b[k]
                + att_rp[e * 3 + 0] * rp_W[k] + att_rp[e * 3 + 1] * rp_W[HIDN + k]
                + att_rp[e * 3 + 2] * rp_W[2 * HIDN + k] + rp_b[k];
        emb[i] = v;
    }
}

// Per-edge attention score (one wave per edge), leaky_relu then dot with attn.
// Lanes 0-15 cover head 0 (k=0..63), lanes 16-31 head 1 (k=64..127).
__global__ void psp_edge_score(const float* __restrict__ f_ni,
                               const float* __restrict__ f_nj,
                               const float* __restrict__ f_fij,
                               const int* __restrict__ src, const int* __restrict__ dst,
                               const float* __restrict__ attn_l,
                               const float* __restrict__ bias_l,
                               float* __restrict__ escore,
                               unsigned* __restrict__ nodemax, int E) {
    int e = blockIdx.x * (blockDim.x >> 5) + (threadIdx.x >> 5);
    if (e >= E) return;
    int lane = threadIdx.x & 31;
    int s = src[e], d = dst[e];
    int k0 = lane << 2;
    float4 a = *(const float4*)(f_ni + (size_t)s * HOUT + k0);
    float4 b = *(const float4*)(f_nj + (size_t)d * HOUT + k0);
    float4 f = *(const float4*)(f_fij + (size_t)e * HOUT + k0);
    float4 bb = *(const float4*)(bias_l + k0);
    float4 at = *(const float4*)(attn_l + k0);
    float p = 0.f, v;
    v = a.x + b.x + f.x + bb.x; v = v > 0.f ? v : NEGSLOPE * v; p += v * at.x;
    v = a.y + b.y + f.y + bb.y; v = v > 0.f ? v : NEGSLOPE * v; p += v * at.y;
    v = a.z + b.z + f.z + bb.z; v = v > 0.f ? v : NEGSLOPE * v; p += v * at.z;
    v = a.w + b.w + f.w + bb.w; v = v > 0.f ? v : NEGSLOPE * v; p += v * at.w;
    for (int off = 1; off < 16; off <<= 1) p += __shfl_xor(p, off, 32);
    if ((lane & 15) == 0) {
        int h = lane >> 4;
        escore[(size_t)e * 2 + h] = p;
        atomicMax(&nodemax[(size_t)d * 2 + h], psp_enc(p));
    }
}

// ex = exp(e - max[dst]); segment-sum of ex into nodesum; escore <- ex in place
__global__ void psp_edge_exp(const int* __restrict__ dst,
                             const unsigned* __restrict__ nodemax,
                             float* __restrict__ escore,
                             float* __restrict__ nodesum, int E) {
    int total = E * 2;
    for (int i = blockIdx.x * blockDim.x + threadIdx.x; i < total;
         i += gridDim.x * blockDim.x) {
        int e = i >> 1, h = i & 1, d = dst[e];
        float m = psp_dec(nodemax[d * 2 + h]);
        unsigned mb = __float_as_uint(m);
        if ((mb & 0x7F800000u) == 0x7F800000u) m = 0.f;   // non-finite -> 0
        float ex = expf(escore[i] - m);
        escore[i] = ex;
        atomicAdd(&nodesum[d * 2 + h], ex);
    }
}

// h_att[dst] += alpha * hn[src]; 128 consecutive lanes share one edge.
__global__ void psp_edge_scatter(const int* __restrict__ src, const int* __restrict__ dst,
                                 const float* __restrict__ escore,
                                 const float* __restrict__ nodesum,
                                 const float* __restrict__ hn,
                                 float* __restrict__ h_att, int E) {
    size_t total = (size_t)E * HOUT;
    size_t stride = (size_t)gridDim.x * blockDim.x;
    for (size_t i = (size_t)blockIdx.x * blockDim.x + threadIdx.x; i < total; i += stride) {
        int e = (int)(i >> 7), k = (int)(i & 127), h = k >> 6;
        int d = dst[e];
        float alpha = escore[(size_t)e * 2 + h] / (nodesum[(size_t)d * 2 + h] + 1e-9f);
        atomicAdd(&h_att[(size_t)d * HOUT + k],
                  alpha * hn[(size_t)src[e] * HOUT + k]);
    }
}

// Max over 6250 nodes per graph, per feature column (coalesced: threads on cols).
__global__ void psp_graph_max(const float* __restrict__ features,
                              unsigned* __restrict__ gemb) {
    int g = blockIdx.y;
    int n0 = blockIdx.x * 256;
    int tid = threadIdx.x;
    float lm0 = -3.402823466e38f, lm1 = -3.402823466e38f;
    int nEnd = (n0 + 256 < NPG) ? n0 + 256 : NPG;
    for (int n = n0; n < nEnd; ++n) {
        const float* row = features + (size_t)(g * NPG + n) * FDIM;
        if (tid < FDIM) lm0 = fmaxf(lm0, row[tid]);
        if (tid + 256 < FDIM) lm1 = fmaxf(lm1, row[tid + 256]);
    }
    if (tid < FDIM) atomicMax(&gemb[g * FDIM + tid], psp_enc(lm0));
    if (tid + 256 < FDIM) atomicMax(&gemb[g * FDIM + tid + 256], psp_enc(lm1));
}

// out[g,n,0:266] = features, out[g,n,266:532] = graph max (decoded)
__global__ void psp_assemble(const float* __restrict__ features,
                             const unsigned* __restrict__ gemb,
                             float* __restrict__ out, int total) {
    for (int i = blockIdx.x * blockDim.x + threadIdx.x; i < total;
         i += gridDim.x * blockDim.x) {
        int g = i / (NPG * 2 * FDIM);
        int r = i - g * (NPG * 2 * FDIM);
        int n = r / (2 * FDIM);
        int c = r - n * (2 * FDIM);
        out[i] = (c < FDIM) ? features[(size_t)(g * NPG + n) * FDIM + c]
                            : psp_dec(gemb[g * FDIM + (c - FDIM)]);
    }
}

static inline int psp_cdiv(long long a, long long b) { return (int)((a + b - 1) / b); }

extern "C" void kernel_launch(void* const* d_in, const int* in_sizes, int n_in,
                              void* d_out, int out_size, void* d_ws, size_t ws_size,
                              hipStream_t stream) {
    const float* feat      = (const float*)d_in[0];
    const float* att_rc    = (const float*)d_in[1];
    const float* att_rp    = (const float*)d_in[2];
    const float* etype_emb = (const float*)d_in[3];
    const float* rid_emb   = (const float*)d_in[4];
    const float* rc_W      = (const float*)d_in[5];
    const float* rc_b      = (const float*)d_in[6];
    const float* rp_W      = (const float*)d_in[7];
    const float* rp_b      = (const float*)d_in[8];
    const float* fe_W0     = (const float*)d_in[9];
    const float* fe_b0     = (const float*)d_in[10];
    const float* fe_W1     = (const float*)d_in[11];
    const float* fe_b1     = (const float*)d_in[12];
    const float* W_ni      = (const float*)d_in[13];
    const float* W_nj      = (const float*)d_in[14];
    const float* W_fij     = (const float*)d_in[15];
    const float* W_node    = (const float*)d_in[16];
    const float* attn      = (const float*)d_in[17];
    const float* egat_bias = (const float*)d_in[18];
    const float* Wm0       = (const float*)d_in[19];
    const float* bm0       = (const float*)d_in[20];
    const float* Wm1       = (const float*)d_in[21];
    const float* bm1       = (const float*)d_in[22];
    const int*   src       = (const int*)d_in[23];
    const int*   dst       = (const int*)d_in[24];
    const int*   etype     = (const int*)d_in[25];
    const int*   rid       = (const int*)d_in[26];

    const int N = in_sizes[0] / INDIM;          // 50000
    const int E = in_sizes[23];                 // 500000

    // ---- workspace carve (256B aligned) ----
    char* wp = (char*)d_ws;
    auto carve = [&](size_t bytes) -> void* {
        void* r = (void*)wp;
        wp += (bytes + 255) & ~(size_t)255;
        return r;
    };
    float*    emb      = (float*)carve((size_t)E * HIDN * 4);
    float*    f_fij    = (float*)carve((size_t)E * HOUT * 4);
    float*    f_ni     = (float*)carve((size_t)N * HOUT * 4);
    float*    f_nj     = (float*)carve((size_t)N * HOUT * 4);
    float*    hn       = (float*)carve((size_t)N * HOUT * 4);
    float*    h_att    = (float*)carve((size_t)N * HOUT * 4);
    float*    mlp_tmp  = (float*)carve((size_t)N * HOUT * 4);
    float*    enc_h    = (float*)carve((size_t)N * HIDN * 4);
    float*    features = (float*)carve((size_t)N * FDIM * 4);
    float*    escore   = (float*)carve((size_t)E * 2 * 4);
    unsigned* nodemax  = (unsigned*)carve((size_t)N * 2 * 4);
    float*    nodesum  = (float*)carve((size_t)N * 2 * 4);
    unsigned* gemb     = (unsigned*)carve((size_t)BGRAPH * FDIM * 4);

    auto gemm = [&](const float* A, int lda, const float* B, int ldb,
                    float* C, int ldc, int M, int K, int Nc,
                    const float* bias, int relu, const float* resid, int rld) {
        int tiles = ((M + 15) / 16) * (Nc / 16);
        int blocks = (tiles + 7) / 8;               // 8 waves (256 threads)/block
        psp_gemm_wmma<<<blocks, 256, 0, stream>>>(A, lda, B, ldb, C, ldc,
                                                  M, K, Nc, bias, relu, resid, rld);
    };

    // features[:,0:10] = feat
    psp_copy_feat<<<psp_cdiv((long long)N * INDIM, 256), 256, 0, stream>>>(feat, features, N);

    // feat encoder: h_enc = relu(feat@W0+b0)@W1+b1 -> features cols 10..73
    gemm(feat, INDIM, fe_W0, HIDN, enc_h, HIDN, N, INDIM, HIDN, fe_b0, 1, nullptr, 0);
    gemm(enc_h, HIDN, fe_W1, HIDN, features + INDIM, FDIM, N, HIDN, HIDN, fe_b1, 0, nullptr, 0);

    // edge embedding
    psp_edge_emb<<<psp_cdiv((long long)E * HIDN, 256), 256, 0, stream>>>(
        att_rc, att_rp, etype_emb, rid_emb, rc_W, rc_b, rp_W, rp_b, etype, rid, emb, E);

    for (int l = 0; l < NLAYERS; ++l) {
        const int colprev = INDIM + l * HIDN;
        const int colnew  = colprev + HIDN;
        const float* h = features + colprev;                 // (N,64) ld=FDIM

        gemm(h, FDIM, W_ni + l * HIDN * HOUT, HOUT, f_ni, HOUT, N, HIDN, HOUT, nullptr, 0, nullptr, 0);
        gemm(h, FDIM, W_nj + l * HIDN * HOUT, HOUT, f_nj, HOUT, N, HIDN, HOUT, nullptr, 0, nullptr, 0);
        gemm(h, FDIM, W_node + l * HIDN * HOUT, HOUT, hn,  HOUT, N, HIDN, HOUT, nullptr, 0, nullptr, 0);
        gemm(emb, HIDN, W_fij + l * HIDN * HOUT, HOUT, f_fij, HOUT, E, HIDN, HOUT, nullptr, 0, nullptr, 0);

        hipMemsetAsync(nodemax, 0, (size_t)N * 2 * 4, stream);   // enc order: 0 < enc(any real)
        hipMemsetAsync(nodesum, 0, (size_t)N * 2 * 4, stream);
        hipMemsetAsync(h_att,   0, (size_t)N * HOUT * 4, stream);

        psp_edge_score<<<psp_cdiv(E, 8), 256, 0, stream>>>(
            f_ni, f_nj, f_fij, src, dst, attn + l * NHEADS * HIDN,
            egat_bias + l * HOUT, escore, nodemax, E);
        psp_edge_exp<<<psp_cdiv((long long)E * 2, 256), 256, 0, stream>>>(
            dst, nodemax, escore, nodesum, E);
        psp_edge_scatter<<<psp_cdiv((long long)E * HOUT, 256), 256, 0, stream>>>(
            src, dst, escore, nodesum, hn, h_att, E);

        // node MLP + residual -> features cols [colnew, colnew+64)
        gemm(h_att, HOUT, Wm0 + l * HOUT * HOUT, HOUT, mlp_tmp, HOUT,
             N, HOUT, HOUT, bm0 + l * HOUT, 1, nullptr, 0);
        gemm(mlp_tmp, HOUT, Wm1 + l * HOUT * HIDN, HIDN, features + colnew, FDIM,
             N, HOUT, HIDN, bm1 + l * HIDN, 0, features + colprev, FDIM);
    }

    // graph max pooling + final assembly
    hipMemsetAsync(gemb, 0, (size_t)BGRAPH * FDIM * 4, stream);
    dim3 gmGrid(psp_cdiv(NPG, 256), BGRAPH);
    psp_graph_max<<<gmGrid, 256, 0, stream>>>(features, gemb);
    psp_assemble<<<psp_cdiv(out_size, 256), 256, 0, stream>>>(
        features, gemb, (float*)d_out, out_size);
}